// NoisyCell_25031069401740
// MI455X (gfx1250) — compile-verified
//
#include <hip/hip_runtime.h>

// ---------------------------------------------------------------------------
// NoisyCell 2-layer tanh RNN over a packed sequence, f32, CDNA5 (gfx1250).
// V_WMMA_F32_16X16X4_F32 keeps full f32 precision across the 256-step chain.
//
// Big-workspace path (if ws fits ~270MB):
//   1) P0 = X @ W_ih0^T + biases        (one parallel 32768x1024x1024 GEMM)
//   2) 256 sequential steps, K=1024:  h0[t] = tanh(P0[t] + h0[t-1]@W_hh0^T)+noise
//   3) P1 = H0_all @ W_ih1^T + biases   (parallel GEMM)
//   4) 256 sequential steps, K=1024 for layer 1
// Fallback path (small ws): fused K=2048 per step (ih + hh in one kernel).
// GEMM inner loops: double-buffered LDS, one barrier per K-chunk,
// global_load_b128 (address_space(1), LOADcnt-only) in flight during WMMA.
// ---------------------------------------------------------------------------

typedef __attribute__((ext_vector_type(2))) float v2f;
typedef __attribute__((ext_vector_type(4))) float v4f;
typedef __attribute__((ext_vector_type(8))) float v8f;

#define GLOBAL_AS __attribute__((address_space(1)))
typedef const GLOBAL_AS float* gcfp;
typedef const GLOBAL_AS v4f*   gcv4p;

constexpr int T = 256;
constexpr int B = 128;
constexpr int E = 1024;
constexpr int H = 1024;
constexpr int L = 2;
constexpr float NOISE_SCALE = 0.1f;

constexpr int LDSP = 20;   // LDS row pitch in floats (80B: 16B-aligned, bank-spread)
constexpr int KC   = 16;   // K chunk staged in LDS

// ---- sequential step kernel: 128 thr = 4 waves, block tile 32x64 -----------
constexpr int SBM = 32;
constexpr int SBN = 64;

// acc = [in0 @ wih^T +] hprev @ whh^T ; val = tanh(acc + (pre ? pre : bias))
//       + 0.1*noise ; hcur = (row < batch_sizes[t]) ? val : hprev
__global__ __launch_bounds__(128)
void rnn_step_kernel(const float* __restrict__ in0,     // [B,1024] (unused if pre)
                     const float* __restrict__ wih,     // [H,1024] (unused if pre)
                     const float* __restrict__ whh,     // [H,H]
                     const float* __restrict__ hprev,   // [B,H]
                     const float* __restrict__ bih,     // [H]      (unused if pre)
                     const float* __restrict__ bhh,     // [H]      (unused if pre)
                     const float* __restrict__ pre,     // [B,H] precomputed ih+bias, or null
                     const float* __restrict__ noise_tl,// [B,H]
                     const int*   __restrict__ batch_sizes,
                     float*       __restrict__ hcur,    // [B,H]
                     int t)
{
    __shared__ __align__(16) float lA[2][SBM * LDSP];
    __shared__ __align__(16) float lB[2][SBN * LDSP];

    const int tid  = threadIdx.x;
    const int lane = tid & 31;
    const int wave = tid >> 5;          // 0..3
    const int wm   = wave >> 1;         // 0..1 -> 16 rows each
    const int wn   = wave & 1;          // 0..1 -> 32 cols each
    const int lg   = lane >> 4;
    const int lm   = lane & 15;

    const int rowBase = blockIdx.y * SBM;
    const int colBase = blockIdx.x * SBN;

    const v8f vzero = {0.f,0.f,0.f,0.f,0.f,0.f,0.f,0.f};
    v8f acc[2]; acc[0] = vzero; acc[1] = vzero;   // 1 m-tile x 2 n-tiles

    // chunk c covers e0 = (c & 63)*KC of phase (c >> 6); pointers pinned to
    // address_space(1) so staging prefetches lower to global_load (LOADcnt only).
    gcfp Aph[2]; gcfp Wph[2];
    int total;
    if (pre) { Aph[0] = (gcfp)hprev; Wph[0] = (gcfp)whh;
               Aph[1] = (gcfp)hprev; Wph[1] = (gcfp)whh; total = 64; }
    else     { Aph[0] = (gcfp)in0;   Wph[0] = (gcfp)wih;
               Aph[1] = (gcfp)hprev; Wph[1] = (gcfp)whh; total = 128; }

    // branch-free staging coordinates: A = 32x16 (128 f4), B = 64x16 (256 f4)
    const int ar  = tid >> 2, aq = tid & 3;   // A rows 0..31
    const int br0 = tid >> 2;                 // B rows 0..31
    const int br1 = br0 + 32;                 // B rows 32..63
    v4f ra, rb0, rb1;

    auto loadChunk = [&](int c) {
        const int ph = c >> 6;
        const int e0 = (c & 63) * KC;
        gcfp Ag = Aph[ph];
        gcfp Wg = Wph[ph];
        ra  = *(gcv4p)(Ag + (size_t)(rowBase + ar ) * 1024 + e0 + aq * 4);
        rb0 = *(gcv4p)(Wg + (size_t)(colBase + br0) * 1024 + e0 + aq * 4);
        rb1 = *(gcv4p)(Wg + (size_t)(colBase + br1) * 1024 + e0 + aq * 4);
    };
    auto storeChunk = [&](int buf) {
        *(v4f*)(&lA[buf][ar  * LDSP + aq * 4]) = ra;
        *(v4f*)(&lB[buf][br0 * LDSP + aq * 4]) = rb0;
        *(v4f*)(&lB[buf][br1 * LDSP + aq * 4]) = rb1;
    };

    loadChunk(0);
    storeChunk(0);
    for (int c = 0; c < total; ++c) {
        const int cur = c & 1;
        const bool more = (c + 1 < total);
        if (more) loadChunk(c + 1);       // global loads in flight during WMMA
        __syncthreads();                  // chunk c fully staged (dscnt only)
        #pragma unroll
        for (int ks = 0; ks < KC; ks += 4) {
            const v2f af  = *(const v2f*)(&lA[cur][(wm * 16 + lm) * LDSP + ks + lg * 2]);
            const v2f bf0 = *(const v2f*)(&lB[cur][(wn * 32 +      lm) * LDSP + ks + lg * 2]);
            const v2f bf1 = *(const v2f*)(&lB[cur][(wn * 32 + 16 + lm) * LDSP + ks + lg * 2]);
            acc[0] = __builtin_amdgcn_wmma_f32_16x16x4_f32(
                false, af, false, bf0, (short)0, acc[0], false, false);
            acc[1] = __builtin_amdgcn_wmma_f32_16x16x4_f32(
                false, af, false, bf1, (short)0, acc[1], false, false);
        }
        if (more) storeChunk(cur ^ 1);    // fill other buffer for next iteration
    }

    const int bs = batch_sizes[t];
    #pragma unroll
    for (int nt = 0; nt < 2; ++nt) {
        const int c = colBase + wn * 32 + nt * 16 + lm;
        const float bias = pre ? 0.f : (bih[c] + bhh[c]);
        #pragma unroll
        for (int v = 0; v < 8; ++v) {
            const int r = rowBase + wm * 16 + lg * 8 + v;
            const size_t off = (size_t)r * H + c;
            const float base = pre ? pre[off] : bias;
            const float val = tanhf(acc[nt][v] + base) + NOISE_SCALE * noise_tl[off];
            const float old = hprev[off];
            hcur[off] = (r < bs) ? val : old;
        }
    }
}

// ---- parallel GEMM: P[r,h] = A[r,:]·W[h,:] + bih[h] + bhh[h] ---------------
// 256 thr = 8 waves (2M x 4N), block tile 128x128, wave tile 64x32 (4x2 tiles)
constexpr int GBM = 128;
constexpr int GBN = 128;

__global__ __launch_bounds__(256)
void gemm_bias_kernel(const float* __restrict__ A,   // [R,1024]
                      const float* __restrict__ W,   // [H,1024]
                      const float* __restrict__ bih,
                      const float* __restrict__ bhh,
                      float* __restrict__ P)         // [R,H]
{
    __shared__ __align__(16) float lA[2][GBM * LDSP];
    __shared__ __align__(16) float lB[2][GBN * LDSP];

    const int tid  = threadIdx.x;
    const int lane = tid & 31;
    const int wave = tid >> 5;          // 0..7
    const int wm   = wave >> 2;         // 0..1 -> 64 rows each
    const int wn   = wave & 3;          // 0..3 -> 32 cols each
    const int lg   = lane >> 4;
    const int lm   = lane & 15;

    const int rowBase = blockIdx.y * GBM;
    const int colBase = blockIdx.x * GBN;

    const v8f vzero = {0.f,0.f,0.f,0.f,0.f,0.f,0.f,0.f};
    v8f acc[4][2];
    #pragma unroll
    for (int mt = 0; mt < 4; ++mt) { acc[mt][0] = vzero; acc[mt][1] = vzero; }

    // staging: A and B tiles are 128x16 = 512 f4 each; 256 threads x 2 each
    const int r0 = tid >> 2, q0 = tid & 3;   // rows 0..63
    const int r1 = r0 + 64;                  // rows 64..127
    gcfp Ag = (gcfp)A;
    gcfp Wg = (gcfp)W;
    v4f ra0, ra1, rb0, rb1;

    auto loadChunk = [&](int c) {
        const int e0 = c * KC;
        ra0 = *(gcv4p)(Ag + (size_t)(rowBase + r0) * 1024 + e0 + q0 * 4);
        ra1 = *(gcv4p)(Ag + (size_t)(rowBase + r1) * 1024 + e0 + q0 * 4);
        rb0 = *(gcv4p)(Wg + (size_t)(colBase + r0) * 1024 + e0 + q0 * 4);
        rb1 = *(gcv4p)(Wg + (size_t)(colBase + r1) * 1024 + e0 + q0 * 4);
    };
    auto storeChunk = [&](int buf) {
        *(v4f*)(&lA[buf][r0 * LDSP + q0 * 4]) = ra0;
        *(v4f*)(&lA[buf][r1 * LDSP + q0 * 4]) = ra1;
        *(v4f*)(&lB[buf][r0 * LDSP + q0 * 4]) = rb0;
        *(v4f*)(&lB[buf][r1 * LDSP + q0 * 4]) = rb1;
    };

    loadChunk(0);
    storeChunk(0);
    for (int c = 0; c < 64; ++c) {
        const int cur = c & 1;
        const bool more = (c + 1 < 64);
        if (more) loadChunk(c + 1);
        __syncthreads();
        #pragma unroll
        for (int ks = 0; ks < KC; ks += 4) {
            v2f af[4], bf[2];
            #pragma unroll
            for (int mt = 0; mt < 4; ++mt)
                af[mt] = *(const v2f*)(&lA[cur][(wm * 64 + mt * 16 + lm) * LDSP + ks + lg * 2]);
            #pragma unroll
            for (int nt = 0; nt < 2; ++nt)
                bf[nt] = *(const v2f*)(&lB[cur][(wn * 32 + nt * 16 + lm) * LDSP + ks + lg * 2]);
            #pragma unroll
            for (int mt = 0; mt < 4; ++mt)
                #pragma unroll
                for (int nt = 0; nt < 2; ++nt)
                    acc[mt][nt] = __builtin_amdgcn_wmma_f32_16x16x4_f32(
                        false, af[mt], false, bf[nt], (short)0, acc[mt][nt], false, false);
        }
        if (more) storeChunk(cur ^ 1);
    }

    #pragma unroll
    for (int mt = 0; mt < 4; ++mt) {
        #pragma unroll
        for (int nt = 0; nt < 2; ++nt) {
            const int c = colBase + wn * 32 + nt * 16 + lm;
            const float bias = bih[c] + bhh[c];
            #pragma unroll
            for (int v = 0; v < 8; ++v) {
                const int r = rowBase + wm * 64 + mt * 16 + lg * 8 + v;
                P[(size_t)r * H + c] = acc[mt][nt][v] + bias;
            }
        }
    }
}

__global__ void init_zero2_kernel(float* __restrict__ a, float* __restrict__ b)
{
    const size_t i = (size_t)blockIdx.x * blockDim.x + threadIdx.x;
    if (i < (size_t)B * H) { a[i] = 0.f; b[i] = 0.f; }
}

__global__ void gather_out_kernel(const float* __restrict__ h0,
                                  const float* __restrict__ h1,
                                  const int* __restrict__ uidx,
                                  float* __restrict__ out)
{
    const int j = blockIdx.x;       // 0..B-1
    const int l = blockIdx.y;       // 0..L-1
    const float* src = l ? h1 : h0;
    const int row = uidx[j];
    for (int h = threadIdx.x; h < H; h += blockDim.x)
        out[((size_t)l * B + j) * H + h] = src[(size_t)row * H + h];
}

extern "C" void kernel_launch(void* const* d_in, const int* in_sizes, int n_in,
                              void* d_out, int out_size, void* d_ws, size_t ws_size,
                              hipStream_t stream)
{
    (void)in_sizes; (void)n_in; (void)out_size;

    const float* x      = (const float*)d_in[0];   // [T,B,E]
    const float* noise  = (const float*)d_in[1];   // [T,L,B,H]
    const float* w_ih0  = (const float*)d_in[2];
    const float* w_hh0  = (const float*)d_in[3];
    const float* b_ih0  = (const float*)d_in[4];
    const float* b_hh0  = (const float*)d_in[5];
    const float* w_ih1  = (const float*)d_in[6];
    const float* w_hh1  = (const float*)d_in[7];
    const float* b_ih1  = (const float*)d_in[8];
    const float* b_hh1  = (const float*)d_in[9];
    const int* batch_sizes = (const int*)d_in[10];
    const int* uidx        = (const int*)d_in[11];
    float* out = (float*)d_out;

    float* ws = (float*)d_ws;
    const size_t BH  = (size_t)B * H;
    const size_t TBH = (size_t)T * BH;

    const dim3 sgrid(H / SBN, B / SBM);            // 16 x 4 = 64 blocks
    const size_t bigNeed = (2 * TBH + 3 * BH) * sizeof(float);   // ~270 MB

    if (ws_size >= bigNeed) {
        // ---------------- big-workspace path ----------------
        float* P     = ws;                 // [T*B, H] precomputed ih terms
        float* H0ALL = ws + TBH;           // [T, B, H] layer-0 history
        float* hz    = ws + 2 * TBH;       // zeroed initial hidden state
        float* h1buf[2] = { hz + BH, hz + 2 * BH };

        init_zero2_kernel<<<dim3((unsigned)((BH + 255) / 256)), 256, 0, stream>>>(hz, h1buf[0]);

        const dim3 ggrid(H / GBN, (T * B) / GBM);  // 8 x 256 = 2048 blocks
        // P0 = X @ w_ih0^T + b_ih0 + b_hh0  (parallel)
        gemm_bias_kernel<<<ggrid, 256, 0, stream>>>(x, w_ih0, b_ih0, b_hh0, P);

        // layer 0: 256 sequential K=1024 steps into the history buffer
        for (int t = 0; t < T; ++t) {
            const float* hp = (t == 0) ? hz : (H0ALL + (size_t)(t - 1) * BH);
            rnn_step_kernel<<<sgrid, 128, 0, stream>>>(
                nullptr, nullptr, w_hh0, hp, nullptr, nullptr,
                P + (size_t)t * BH, noise + ((size_t)t * L + 0) * BH,
                batch_sizes, H0ALL + (size_t)t * BH, t);
        }

        // P1 = H0_all @ w_ih1^T + b_ih1 + b_hh1  (parallel, overwrites P)
        gemm_bias_kernel<<<ggrid, 256, 0, stream>>>(H0ALL, w_ih1, b_ih1, b_hh1, P);

        // layer 1: 256 sequential K=1024 steps, ping-pong state
        for (int t = 0; t < T; ++t) {
            const int pp = t & 1, cp = pp ^ 1;
            rnn_step_kernel<<<sgrid, 128, 0, stream>>>(
                nullptr, nullptr, w_hh1, h1buf[pp], nullptr, nullptr,
                P + (size_t)t * BH, noise + ((size_t)t * L + 1) * BH,
                batch_sizes, h1buf[cp], t);
        }

        gather_out_kernel<<<dim3(B, L), 256, 0, stream>>>(
            H0ALL + (size_t)(T - 1) * BH, h1buf[0], uidx, out);
    } else {
        // ---------------- fallback: fused K=2048 per step (ws = 2 MB) -------
        float* h0buf[2] = { ws + 0 * BH, ws + 1 * BH };
        float* h1buf[2] = { ws + 2 * BH, ws + 3 * BH };

        init_zero2_kernel<<<dim3((unsigned)((BH + 255) / 256)), 256, 0, stream>>>(h0buf[0], h1buf[0]);

        for (int t = 0; t < T; ++t) {
            const int pp = t & 1, cp = pp ^ 1;
            rnn_step_kernel<<<sgrid, 128, 0, stream>>>(
                x + (size_t)t * B * E, w_ih0, w_hh0, h0buf[pp], b_ih0, b_hh0,
                nullptr, noise + ((size_t)t * L + 0) * BH, batch_sizes, h0buf[cp], t);
            rnn_step_kernel<<<sgrid, 128, 0, stream>>>(
                h0buf[cp], w_ih1, w_hh1, h1buf[pp], b_ih1, b_hh1,
                nullptr, noise + ((size_t)t * L + 1) * BH, batch_sizes, h1buf[cp], t);
        }

        gather_out_kernel<<<dim3(B, L), 256, 0, stream>>>(h0buf[0], h1buf[0], uidx, out);
    }
}